// SpatialThresholdSelector_41635412967566
// MI455X (gfx1250) — compile-verified
//
#include <hip/hip_runtime.h>
#include <hip/hip_bf16.h>

typedef __attribute__((ext_vector_type(2))) float v2f;
typedef __attribute__((ext_vector_type(8))) float v8f;

#define BATCH      32
#define NPATCH     4096
#define DMODEL     768
#define IMG        1024
#define IMGPIX     (IMG*IMG)
#define SIDE       64
#define KSEL       1024
#define COG_CHUNKS 256
#define THRESH_F   0.3f

// ---------------------------------------------------------------------------
// K1: per-batch partial sums of (y*v, x*v, v) over the 1024x1024 line drawing.
// 128 MB streamed with float4 loads; deterministic block-tree reduction.
// ---------------------------------------------------------------------------
__global__ __launch_bounds__(256)
void cog_partial_kernel(const float* __restrict__ ld, float4* __restrict__ partials) {
    const int b = blockIdx.y, chunk = blockIdx.x, t = threadIdx.x;
    const float4* img = (const float4*)(ld + (size_t)b * IMGPIX);
    float sy = 0.f, sx = 0.f, sm = 0.f;
    const int base4 = chunk * 1024;              // 4096 pixels per chunk
#pragma unroll
    for (int q = 0; q < 4; ++q) {
        int idx4 = base4 + t + 256 * q;
        float4 v = img[idx4];
        int p = idx4 * 4;                        // pixel index; 4 px share one row
        float r = (float)(p >> 10);              // row (W=1024)
        float c = (float)(p & 1023);             // base column
        float s = v.x + v.y + v.z + v.w;
        sm += s;
        sy += r * s;
        sx += c * s + (v.y + 2.f * v.z + 3.f * v.w);
    }
    __shared__ float ry[256], rx[256], rm[256];
    ry[t] = sy; rx[t] = sx; rm[t] = sm;
    __syncthreads();
    for (int off = 128; off > 0; off >>= 1) {
        if (t < off) { ry[t] += ry[t+off]; rx[t] += rx[t+off]; rm[t] += rm[t+off]; }
        __syncthreads();
    }
    if (t == 0)   // linspace(0,1,1024) step = 1/1023
        partials[b * COG_CHUNKS + chunk] =
            make_float4(ry[0] * (1.f/1023.f), rx[0] * (1.f/1023.f), rm[0], 0.f);
}

// ---------------------------------------------------------------------------
// K2: reduce 256 partials per batch -> (cog_y, cog_x).
// ---------------------------------------------------------------------------
__global__ __launch_bounds__(256)
void cog_final_kernel(const float4* __restrict__ partials, float* __restrict__ cog) {
    const int b = blockIdx.x, t = threadIdx.x;
    float4 v = partials[b * COG_CHUNKS + t];
    __shared__ float ry[256], rx[256], rm[256];
    ry[t] = v.x; rx[t] = v.y; rm[t] = v.z;
    __syncthreads();
    for (int off = 128; off > 0; off >>= 1) {
        if (t < off) { ry[t] += ry[t+off]; rx[t] += rx[t+off]; rm[t] += rm[t+off]; }
        __syncthreads();
    }
    if (t == 0) {
        float tm = rm[0] + 1e-8f;
        cog[2*b + 0] = ry[0] / tm;   // cog_y
        cog[2*b + 1] = rx[0] / tm;   // cog_x
    }
}

// ---------------------------------------------------------------------------
// K3: squared distances as an exact K=4 GEMM on V_WMMA_F32_16X16X4_F32.
//   d2 = (gy-cy)^2 + (gx-cx)^2 = [gy,gx,gy^2+gx^2,1] . [-2cy,-2cx,1,cy^2+cx^2]
// A(16x4): lanes 0-15 hold K=0,1 in VGPR0/1; lanes 16-31 hold K=2,3.
// B(4x16): VGPR0 = rows K0(lanes 0-15)/K2(16-31), VGPR1 = K1/K3.
// D(16x16): VGPR r = M=r (lanes 0-15), M=r+8 (lanes 16-31), N = lane&15.
// One wave = 16 grid points x 16 batches; 256 mtiles x 2 ntiles = 512 waves.
// ---------------------------------------------------------------------------
__global__ __launch_bounds__(128)
void gaussian_wmma_kernel(const float* __restrict__ scores,
                          const float* __restrict__ cog,
                          float* __restrict__ wsw) {
    const int wid   = blockIdx.x * 4 + (threadIdx.x >> 5);
    const int lane  = threadIdx.x & 31;
    const int mtile = wid >> 1;          // 0..255 (grid-point tiles)
    const int ntile = wid & 1;           // 0..1   (batch tiles)
    const int half  = lane >> 4;
    const int l15   = lane & 15;

    const int gp = mtile * 16 + l15;     // grid point for A row
    const float gy = (float)(gp >> 6) * (1.f/63.f);   // linspace(0,1,64)
    const float gx = (float)(gp & 63) * (1.f/63.f);
    const int bt = ntile * 16 + l15;     // batch for B column
    const float cy = cog[2*bt + 0];
    const float cx = cog[2*bt + 1];

    v2f a, bv;
    if (half == 0) { a.x = gy;              a.y = gx;
                     bv.x = -2.f * cy;      bv.y = -2.f * cx; }
    else           { a.x = gy*gy + gx*gx;   a.y = 1.f;
                     bv.x = 1.f;            bv.y = cy*cy + cx*cx; }

    v8f c = {};
    c = __builtin_amdgcn_wmma_f32_16x16x4_f32(false, a, false, bv,
                                              (short)0, c, false, false);
#pragma unroll
    for (int r = 0; r < 8; ++r) {
        int gpo   = mtile * 16 + (r + half * 8);   // D-matrix row (grid point)
        int batch = ntile * 16 + l15;              // D-matrix col (batch)
        float d2  = c[r];
        // gw = exp(-d2 / (2 * 0.25^2)) = exp(-8 * d2); weighted = scores * gw
        wsw[batch * NPATCH + gpo] =
            scores[batch * NPATCH + gpo] * __expf(-8.f * d2);
    }
}

// ---------------------------------------------------------------------------
// K4: exact reproduction of _select_indices via rank-by-counting on distinct
// 64-bit keys  key = (~ordered(w) << 12) | index  (descending w, index tiebreak).
//   rank[i]           = #(key_j < key_i)
//   above             = key < KT (strict w > 0.3)
//   partial && above  -> pos = prefix-count of above by index
//   otherwise         -> pos = rank        (non-above ranks already >= count)
// Positions are collision-free; scatter sel[pos] = i for pos < k.
// Grid: 32 batches x 16 chunks of 256 elements; keys live in 32 KB LDS.
// ---------------------------------------------------------------------------
__global__ __launch_bounds__(256)
void select_kernel(const float* __restrict__ wsw, int* __restrict__ sel) {
    const int b = blockIdx.y, ch = blockIdx.x, t = threadIdx.x;
    __shared__ unsigned long long skey[NPATCH];      // 32 KB
    const float* w = wsw + (size_t)b * NPATCH;
#pragma unroll
    for (int q = 0; q < NPATCH / 256; ++q) {
        int j = t + 256 * q;
        unsigned u  = __float_as_uint(w[j]);
        unsigned au = (u & 0x80000000u) ? ~u : (u | 0x80000000u);  // ascending
        skey[j] = ((unsigned long long)(~au) << 12) | (unsigned)j; // descending
    }
    __syncthreads();

    const unsigned ut = __float_as_uint(THRESH_F) | 0x80000000u;
    const unsigned long long KT = ((unsigned long long)(~ut) << 12);

    const int i = ch * 256 + t;
    const unsigned long long Ki = skey[i];
    unsigned r = 0, pa = 0, ca = 0;
#pragma unroll 8
    for (int j = 0; j < NPATCH; ++j) {
        unsigned long long kj = skey[j];
        unsigned ab = (kj < KT) ? 1u : 0u;
        r  += (kj < Ki) ? 1u : 0u;
        ca += ab;
        pa += ab & ((j < i) ? 1u : 0u);
    }
    const bool partial = (ca > 0u) && (ca < (unsigned)KSEL);
    const bool above   = (Ki < KT);
    const unsigned pos = (partial && above) ? pa : r;
    if (pos < (unsigned)KSEL) sel[b * KSEL + pos] = i;
}

// ---------------------------------------------------------------------------
// K5: gather selected patch rows + positional embedding rows, float4 wide.
// One block per (batch, slot): 192 lanes x float4 = 768 floats.
// pos table (12.6 MB) stays L2-resident and is reused 32x.
// ---------------------------------------------------------------------------
__global__ __launch_bounds__(192)
void gather_kernel(const float* __restrict__ patches,
                   const float* __restrict__ posemb,
                   const int* __restrict__ sel,
                   float* __restrict__ out) {
    const int b = blockIdx.y, j = blockIdx.x, t = threadIdx.x;
    const int row = sel[b * KSEL + j];
    const float4* src = (const float4*)(patches + ((size_t)b * NPATCH + row) * DMODEL);
    const float4* pe  = (const float4*)(posemb  + (size_t)(row + 1) * DMODEL);
    float4*       dst = (float4*)(out + ((size_t)b * KSEL + j) * DMODEL);
    float4 v = src[t], p = pe[t];
    dst[t] = make_float4(v.x + p.x, v.y + p.y, v.z + p.z, v.w + p.w);
}

// ---------------------------------------------------------------------------
extern "C" void kernel_launch(void* const* d_in, const int* in_sizes, int n_in,
                              void* d_out, int out_size, void* d_ws, size_t ws_size,
                              hipStream_t stream) {
    const float* patches = (const float*)d_in[0];   // (32, 4096, 768) f32
    const float* posemb  = (const float*)d_in[1];   // (1, 4097, 768)  f32
    const float* scores  = (const float*)d_in[2];   // (32, 4096)      f32
    const float* ld      = (const float*)d_in[3];   // (32, 1, 1024, 1024) f32
    float* out = (float*)d_out;                     // (32, 1024, 768) f32

    // workspace layout (floats): < 1 MB total
    float*  ws       = (float*)d_ws;
    float4* partials = (float4*)ws;                         // 32*256 float4
    float*  cog      = ws + BATCH * COG_CHUNKS * 4;         // 64 floats
    float*  wsw      = cog + BATCH * 2;                     // 32*4096 floats
    int*    sel      = (int*)(wsw + BATCH * NPATCH);        // 32*1024 ints

    dim3 g1(COG_CHUNKS, BATCH);
    cog_partial_kernel<<<g1, 256, 0, stream>>>(ld, partials);
    cog_final_kernel<<<BATCH, 256, 0, stream>>>(partials, cog);
    gaussian_wmma_kernel<<<128, 128, 0, stream>>>(scores, cog, wsw);
    dim3 g4(NPATCH / 256, BATCH);
    select_kernel<<<g4, 256, 0, stream>>>(wsw, sel);
    dim3 g5(KSEL, BATCH);
    gather_kernel<<<g5, 192, 0, stream>>>(patches, posemb, sel, out);
}